// SupConLoss_74234214744331
// MI455X (gfx1250) — compile-verified
//
#include <hip/hip_runtime.h>

typedef float v2f __attribute__((ext_vector_type(2)));
typedef float v8f __attribute__((ext_vector_type(8)));

#define M_ROWS    8192
#define K_DIM     128
#define INV_T     5.0f                 // 1 / TEMPERATURE
#define NTILE     32                   // columns staged per iteration (two 16x16 C tiles)
#define NITER     (M_ROWS / NTILE)     // 256
#define LDS_PITCH 132                  // 128 + 4 pad -> conflict-free ds_load_b64
#define WAVES_PB  4                    // 4 waves * 16 rows = 64 rows per block
#define NBLOCKS   (M_ROWS / (WAVES_PB * 16))  // 128 blocks
#define NWAVES    (NBLOCKS * WAVES_PB)        // 512 partials

__global__ __launch_bounds__(WAVES_PB * 32) void supcon_fused_kernel(
    const float* __restrict__ f,      // [8192, 128] fp32
    const int*   __restrict__ labels, // [8192]
    float*       __restrict__ ws)     // [512] per-wave partial loss sums
{
    __shared__ float ldsB[2][NTILE * LDS_PITCH];  // double-buffered 32x128 tile
    __shared__ int   ldsLab[2][NTILE];

    const int tid  = threadIdx.x;
    const int lane = tid & 31;
    const int wave = tid >> 5;
    const int n    = lane & 15;          // M-slot (A) / N-slot (B,C)
    const int kb   = (lane >> 4) * 2;    // K phase 0 or 2 (ISA 16x4 f32 layout)
    const int r0   = (blockIdx.x * WAVES_PB + wave) * 16;

    // ---- Preload & prescale A fragments (16 rows x 128 K -> 32 frags) ----
    v2f a[32];
    {
        const float* arow = f + (size_t)(r0 + n) * K_DIM;
        #pragma unroll
        for (int t = 0; t < 32; ++t) {
            a[t].x = arow[4 * t + kb]     * INV_T;
            a[t].y = arow[4 * t + kb + 1] * INV_T;
        }
    }

    // Labels of the 8 C-rows this lane owns (rows 0-7 lanes 0-15, 8-15 lanes 16-31).
    const int rowBase = r0 + ((lane >> 4) << 3);
    int rlab[8];
    #pragma unroll
    for (int v = 0; v < 8; ++v) rlab[v] = labels[rowBase + v];

    // Two column slots per lane (n and n+16): stats[0..7] slot0, [8..15] slot1.
    float mrun[16], drun[16], ssum[16], cnt[16];
    #pragma unroll
    for (int i = 0; i < 16; ++i) {
        mrun[i] = -3.0e38f; drun[i] = 0.f; ssum[i] = 0.f; cnt[i] = 0.f;
    }

    // Cooperative stage-in mapping: 32 cols x 128 K = 4096 floats / 128 thr = 32 each.
    const int loadCol = tid >> 2;          // 0..31
    const int loadK   = (tid & 3) * 32;    // 0,32,64,96

    float4 pf[8];
    int labPf = 0;

    // ---- Prologue: fetch + stage tile 0 ----
    {
        const float4* gsrc = (const float4*)(f + (size_t)loadCol * K_DIM + loadK);
        #pragma unroll
        for (int i = 0; i < 8; ++i) pf[i] = gsrc[i];
        if (tid < NTILE) labPf = labels[tid];

        float* dst = &ldsB[0][loadCol * LDS_PITCH + loadK];
        #pragma unroll
        for (int i = 0; i < 8; ++i) *(float4*)(dst + 4 * i) = pf[i];
        if (tid < NTILE) ldsLab[0][tid] = labPf;
    }
    __syncthreads();

    for (int ct = 0; ct < NITER; ++ct) {
        const int buf = ct & 1;
        const int c0  = ct * NTILE;

        // ---- Issue prefetch of tile ct+1 into registers (latency hidden) ----
        if (ct + 1 < NITER) {
            const float4* gsrc =
                (const float4*)(f + (size_t)(c0 + NTILE + loadCol) * K_DIM + loadK);
            #pragma unroll
            for (int i = 0; i < 8; ++i) pf[i] = gsrc[i];
            if (tid < NTILE) labPf = labels[c0 + NTILE + tid];
        }

        // ---- Two 16x16 sim tiles: 64 x V_WMMA_F32_16X16X4_F32, shared A ----
        const float* bbase = &ldsB[buf][0];
        v8f c8a = {}, c8b = {};
        #pragma unroll
        for (int t = 0; t < 32; ++t) {
            v2f b0 = *(const v2f*)&bbase[ n        * LDS_PITCH + 4 * t + kb];
            v2f b1 = *(const v2f*)&bbase[(n + 16) * LDS_PITCH + 4 * t + kb];
            c8a = __builtin_amdgcn_wmma_f32_16x16x4_f32(
                      false, a[t], false, b0, (short)0, c8a, false, false);
            c8b = __builtin_amdgcn_wmma_f32_16x16x4_f32(
                      false, a[t], false, b1, (short)0, c8b, false, false);
        }

        // ---- Fused online log-sum-exp + label-masked sums (1 exp / element) ----
        const int lab0 = ldsLab[buf][n];
        const int lab1 = ldsLab[buf][n + 16];
        const int dv0  = (c0 + n) - rowBase;   // diagonal when v == dv0
        const int dv1  = dv0 + 16;

        #pragma unroll
        for (int v = 0; v < 8; ++v) {
            {   // slot 0 (column c0 + n)
                float x = c8a[v];
                bool  diag = (v == dv0);
                float m = mrun[v], d = drun[v];
                bool  gt = x > m;
                float e  = __expf(-fabsf(x - m));        // exp(-|x-m|)
                float ad = diag ? 0.f : (gt ? 1.f : e);
                drun[v] = (gt ? d * e : d) + ad;
                mrun[v] = gt ? x : m;
                bool sl = (lab0 == rlab[v]) && !diag;
                ssum[v] += sl ? x : 0.f;
                cnt[v]  += sl ? 1.f : 0.f;
            }
            {   // slot 1 (column c0 + 16 + n)
                float x = c8b[v];
                bool  diag = (v == dv1);
                float m = mrun[8 + v], d = drun[8 + v];
                bool  gt = x > m;
                float e  = __expf(-fabsf(x - m));
                float ad = diag ? 0.f : (gt ? 1.f : e);
                drun[8 + v] = (gt ? d * e : d) + ad;
                mrun[8 + v] = gt ? x : m;
                bool sl = (lab1 == rlab[v]) && !diag;
                ssum[8 + v] += sl ? x : 0.f;
                cnt[8 + v]  += sl ? 1.f : 0.f;
            }
        }

        // ---- Stage prefetched tile into the other buffer ----
        if (ct + 1 < NITER) {
            float* dst = &ldsB[buf ^ 1][loadCol * LDS_PITCH + loadK];
            #pragma unroll
            for (int i = 0; i < 8; ++i) *(float4*)(dst + 4 * i) = pf[i];
            if (tid < NTILE) ldsLab[buf ^ 1][tid] = labPf;
        }
        __syncthreads();
    }

    // ---- Reduction: merge the lane's two slots, then 16-lane butterfly ----
    float acc = 0.f;
    #pragma unroll
    for (int v = 0; v < 8; ++v) {
        float m1 = mrun[v],     d1 = drun[v];
        float m2 = mrun[8 + v], d2 = drun[8 + v];
        float m  = fmaxf(m1, m2);
        float d  = d1 * __expf(m1 - m) + d2 * __expf(m2 - m);
        float s  = ssum[v] + ssum[8 + v];
        float c  = cnt[v]  + cnt[8 + v];
        #pragma unroll
        for (int msk = 1; msk < 16; msk <<= 1) {
            float om = __shfl_xor(m, msk, 32);
            float od = __shfl_xor(d, msk, 32);
            float os = __shfl_xor(s, msk, 32);
            float oc = __shfl_xor(c, msk, 32);
            float nm = fmaxf(m, om);
            d = d * __expf(m - nm) + od * __expf(om - nm);
            m = nm; s += os; c += oc;
        }
        // loss_i = -((s - c*m) - c*log d) / (c + 1)  (diag included in denominator)
        acc += -((s - c * m) - c * __logf(d)) / (c + 1.0f);
    }
    acc += __shfl_xor(acc, 16, 32);   // combine row-halves
    if (lane == 0) ws[blockIdx.x * WAVES_PB + wave] = acc;   // deterministic partial
}

// Fixed-order tree reduction of the 512 per-wave partials -> bitwise deterministic.
__global__ __launch_bounds__(256) void supcon_reduce_kernel(
    const float* __restrict__ ws, float* __restrict__ out)
{
    __shared__ float sm[256];
    const int tid = threadIdx.x;
    sm[tid] = ws[tid] + ws[tid + 256];
    __syncthreads();
    #pragma unroll
    for (int s = 128; s > 0; s >>= 1) {
        if (tid < s) sm[tid] += sm[tid + s];
        __syncthreads();
    }
    if (tid == 0) out[0] = sm[0] * (1.0f / (float)M_ROWS);
}

extern "C" void kernel_launch(void* const* d_in, const int* in_sizes, int n_in,
                              void* d_out, int out_size, void* d_ws, size_t ws_size,
                              hipStream_t stream) {
    (void)in_sizes; (void)n_in; (void)out_size; (void)ws_size;
    const float* f      = (const float*)d_in[0];
    const int*   labels = (const int*)d_in[1];
    float*       out    = (float*)d_out;
    float*       ws     = (float*)d_ws;   // 512 floats of scratch

    supcon_fused_kernel<<<NBLOCKS, WAVES_PB * 32, 0, stream>>>(f, labels, ws);
    supcon_reduce_kernel<<<1, 256, 0, stream>>>(ws, out);
}